// GenCTBE_22127671509386
// MI455X (gfx1250) — compile-verified
//
#include <hip/hip_runtime.h>
#include <hip/hip_bf16.h>

// Problem constants (from reference)
#define BATCH 4096
#define U_DIM 256
#define D_DIM 64
#define IN_DIM 320      // U_DIM + D_DIM
#define H_DIM 8192
#define Y_DIM 256
#define ORDER 4096      // H_DIM/2
#define DELTA 0.1f

typedef float v2f __attribute__((ext_vector_type(2)));
typedef float v8f __attribute__((ext_vector_type(8)));

// ---------------------------------------------------------------------------
// 1) rec[h] = rotation of h_init state (single vector, [8192])
// ---------------------------------------------------------------------------
__global__ void prep_rec_kernel(const float* __restrict__ h,
                                const float* __restrict__ omega,
                                float* __restrict__ rec) {
  int k = blockIdx.x * 256 + threadIdx.x;
  if (k < ORDER) {
    float s, c;
    sincosf(omega[k] * DELTA, &s, &c);
    float h0 = h[2 * k], h1 = h[2 * k + 1];
    rec[2 * k]     =  c * h0 + s * h1;
    rec[2 * k + 1] = -s * h0 + c * h1;
  }
}

// ---------------------------------------------------------------------------
// 2) CT[h][j] = C_eff^T : transpose of C_w with the 2x2 pair-mix folded in.
//    C_eff[j,2k]   = ( sin*C[j,2k] + (cos-1)*C[j,2k+1]) / omega
//    C_eff[j,2k+1] = (-(cos-1)*C[j,2k] + sin*C[j,2k+1]) / omega
//    LDS-tiled 32x32 transpose: coalesced loads and stores.
// ---------------------------------------------------------------------------
__global__ __launch_bounds__(256) void ceffT_kernel(const float* __restrict__ Cw,
                                                    const float* __restrict__ omega,
                                                    float* __restrict__ CT) {
  __shared__ float tile[32][33];
  int h0 = blockIdx.x * 32;          // h-tile (0..255)
  int j0 = blockIdx.y * 32;          // j-tile (0..7)
  int tx = threadIdx.x & 31;
  int ty = threadIdx.x >> 5;         // 0..7
#pragma unroll
  for (int r = 0; r < 4; ++r) {      // tile[j_local][h_local]
    int jl = ty + 8 * r;
    tile[jl][tx] = Cw[(size_t)(j0 + jl) * H_DIM + h0 + tx];
  }
  __syncthreads();
#pragma unroll
  for (int r = 0; r < 4; ++r) {
    int lh = ty + 8 * r;             // h_local for the store
    int gh = h0 + lh;
    int k  = gh >> 1;
    float s, c;
    float om = omega[k];
    sincosf(om * DELTA, &s, &c);
    float io  = 1.0f / om;
    float cm1 = c - 1.0f;
    int lp = lh & ~1;                // even pair base (pair stays inside tile)
    float c0 = tile[tx][lp];
    float c1 = tile[tx][lp + 1];
    float v = (lh & 1) ? (-cm1 * c0 + s * c1) * io
                       : (   s * c0 + cm1 * c1) * io;
    CT[(size_t)gh * Y_DIM + j0 + tx] = v;   // coalesced in tx
  }
}

// ---------------------------------------------------------------------------
// 3) bias[j] = rec . C_w[j,:]   (one block per output, LDS tree reduce)
// ---------------------------------------------------------------------------
__global__ __launch_bounds__(256) void bias_kernel(const float* __restrict__ rec,
                                                   const float* __restrict__ Cw,
                                                   float* __restrict__ bias) {
  __shared__ float red[256];
  int j = blockIdx.x;
  int t = threadIdx.x;
  const float* row = Cw + (size_t)j * H_DIM;
  float acc = 0.f;
  for (int h = t; h < H_DIM; h += 256) acc += rec[h] * row[h];
  red[t] = acc;
  __syncthreads();
  for (int s = 128; s > 0; s >>= 1) {
    if (t < s) red[t] += red[t + s];
    __syncthreads();
  }
  if (t == 0) bias[j] = red[0];
}

// ---------------------------------------------------------------------------
// 4) Mpart[ks] = C_eff[256,8192(slice)] @ B_w[8192(slice),320]
//    f32 WMMA 16x16x4, one wave per (i-tile, j-tile, k-slice). Split-K = 8
//    (partial buffers, deterministic — no float atomics).
// ---------------------------------------------------------------------------
__global__ __launch_bounds__(32) void gemm_M_kernel(const float* __restrict__ CT,
                                                    const float* __restrict__ Bw,
                                                    float* __restrict__ Mpart) {
  int it = blockIdx.x;               // 0..19  (i0 in 0..304)
  int jt = blockIdx.y;               // 0..15
  int ks = blockIdx.z;               // 0..7
  int lane = threadIdx.x;
  int half = lane >> 4;              // 0: K={0,1}, 1: K={2,3}
  int l16  = lane & 15;
  int j0 = jt * 16, i0 = it * 16;
  int kofs = half * 2;

  const float* ctp = CT + j0 + l16;  // A: C_eff[j0+l16][k]  (h-major => coalesced)
  const float* bwp = Bw + i0 + l16;  // B: B_w[k][i0+l16]    (coalesced)
  v8f acc = {};

  int kbeg = ks * (H_DIM / 8);
  int kend = kbeg + (H_DIM / 8);
#pragma unroll 4
  for (int k = kbeg; k < kend; k += 4) {
    v2f a, b;
    a.x = ctp[(size_t)(k + kofs) * Y_DIM];
    a.y = ctp[(size_t)(k + kofs + 1) * Y_DIM];
    b.x = bwp[(size_t)(k + kofs) * IN_DIM];
    b.y = bwp[(size_t)(k + kofs + 1) * IN_DIM];
    acc = __builtin_amdgcn_wmma_f32_16x16x4_f32(false, a, false, b,
                                                (short)0, acc, false, false);
  }
  float* mp = Mpart + (size_t)ks * (Y_DIM * IN_DIM);
#pragma unroll
  for (int r = 0; r < 8; ++r) {      // C/D layout: row = r + 8*half, col = l16
    mp[(size_t)(j0 + r + half * 8) * IN_DIM + i0 + l16] = acc[r];
  }
}

// ---------------------------------------------------------------------------
// 5) M = sum of 8 split-K partials
// ---------------------------------------------------------------------------
__global__ void reduceM_kernel(const float* __restrict__ Mpart,
                               float* __restrict__ M) {
  int idx = blockIdx.x * 256 + threadIdx.x;
  if (idx < Y_DIM * IN_DIM) {
    float s = 0.f;
#pragma unroll
    for (int p = 0; p < 8; ++p) s += Mpart[(size_t)p * (Y_DIM * IN_DIM) + idx];
    M[idx] = s;
  }
}

// ---------------------------------------------------------------------------
// 6) y[b][j] = bias[j] + sum_c udu[b][c] * M[j][c]
//    udu = concat(du, u) handled in the load (K tiles never straddle c=64).
//    float4 K-loads: one global_load_b128 per lane per K-step; lanes 0-15 use
//    .xy (K=0,1), lanes 16-31 use .zw (K=2,3) per the 16x4 f32 A layout.
// ---------------------------------------------------------------------------
__global__ __launch_bounds__(128) void gemm_y_kernel(const float* __restrict__ u,
                                                     const float* __restrict__ du,
                                                     const float* __restrict__ M,
                                                     const float* __restrict__ bias,
                                                     float* __restrict__ out) {
  int wave = threadIdx.x >> 5;
  int lane = threadIdx.x & 31;
  int half = lane >> 4;
  int l16  = lane & 15;
  int jt = blockIdx.x * 4 + wave;    // 0..15
  int bt = blockIdx.y;               // 0..255
  int b0 = bt * 16, j0 = jt * 16;

  const float* durow = du + (size_t)(b0 + l16) * D_DIM;
  const float* urow  = u  + (size_t)(b0 + l16) * U_DIM;
  const float* mrow  = M  + (size_t)(j0 + l16) * IN_DIM;

  v8f acc = {};
#pragma unroll 4
  for (int kc = 0; kc < IN_DIM; kc += 4) {
    const float* ap = (kc < D_DIM) ? (durow + kc) : (urow + (kc - D_DIM));
    float4 qa = *(const float4*)ap;
    float4 qb = *(const float4*)(mrow + kc);
    v2f a, b;
    a.x = half ? qa.z : qa.x;  a.y = half ? qa.w : qa.y;
    b.x = half ? qb.z : qb.x;  b.y = half ? qb.w : qb.y;
    acc = __builtin_amdgcn_wmma_f32_16x16x4_f32(false, a, false, b,
                                                (short)0, acc, false, false);
  }
  float bj = bias[j0 + l16];
#pragma unroll
  for (int r = 0; r < 8; ++r) {
    out[(size_t)(b0 + r + half * 8) * Y_DIM + j0 + l16] = acc[r] + bj;
  }
}

// ---------------------------------------------------------------------------
extern "C" void kernel_launch(void* const* d_in, const int* in_sizes, int n_in,
                              void* d_out, int out_size, void* d_ws, size_t ws_size,
                              hipStream_t stream) {
  const float* u     = (const float*)d_in[0];   // [4096,256]
  const float* du    = (const float*)d_in[1];   // [4096,64]
  const float* h     = (const float*)d_in[2];   // [1,8192]
  const float* omega = (const float*)d_in[3];   // [4096]
  const float* Bw    = (const float*)d_in[4];   // [8192,320]
  const float* Cw    = (const float*)d_in[5];   // [256,8192]
  float* y = (float*)d_out;                     // [4096,256]

  // Workspace layout (floats): rec | bias | CT | M | Mpart  (~11.4 MB total)
  float* w     = (float*)d_ws;
  float* rec   = w;                                   // 8192
  float* bias  = w + H_DIM;                           // 256
  float* CT    = w + H_DIM + Y_DIM;                   // 8192*256 (16B aligned)
  float* M     = CT + (size_t)H_DIM * Y_DIM;          // 256*320
  float* Mpart = M + (size_t)Y_DIM * IN_DIM;          // 8*256*320

  prep_rec_kernel<<<ORDER / 256, 256, 0, stream>>>(h, omega, rec);
  ceffT_kernel<<<dim3(H_DIM / 32, Y_DIM / 32), 256, 0, stream>>>(Cw, omega, CT);
  bias_kernel<<<Y_DIM, 256, 0, stream>>>(rec, Cw, bias);
  gemm_M_kernel<<<dim3(IN_DIM / 16, Y_DIM / 16, 8), 32, 0, stream>>>(CT, Bw, Mpart);
  reduceM_kernel<<<(Y_DIM * IN_DIM + 255) / 256, 256, 0, stream>>>(Mpart, M);
  gemm_y_kernel<<<dim3(Y_DIM / 16 / 4, BATCH / 16), 128, 0, stream>>>(u, du, M, bias, y);
}